// PaiNN_7206955123274
// MI455X (gfx1250) — compile-verified
//
#include <hip/hip_runtime.h>
#include <math.h>

#define F 128
#define NRBF 20
#define CUTOFF 5.0f
#define EPS 1e-8f
#define PI_F 3.14159265358979323846f
#define LDS_PAD 4

typedef float v2f __attribute__((ext_vector_type(2)));
typedef float v8f __attribute__((ext_vector_type(8)));
typedef int   v4i __attribute__((ext_vector_type(4)));

#if defined(__has_builtin)
#if __has_builtin(__builtin_amdgcn_global_load_async_to_lds_b128)
#define HAVE_ASYNC_LDS 1
#endif
#endif
#ifndef HAVE_ASYNC_LDS
#define HAVE_ASYNC_LDS 0
#endif

// address-space-qualified v4i for the async-to-LDS builtin
typedef __attribute__((address_space(1))) v4i gv4i;   // global
typedef __attribute__((address_space(3))) v4i lv4i;   // LDS

// ---------------------------------------------------------------------------
// Fused GEMM + bias (+ optional SiLU) using V_WMMA_F32_16X16X4_F32.
// out[n][o] = act( sum_k in[n][k] * W[o][k] + b[o] ).
// One 256-thread block (8 waves) per 16-row x 128-col output block:
//  - 16xK A-strip staged once into LDS via GLOBAL_LOAD_ASYNC_TO_LDS_B128
//  - each wave owns one 16x16 tile: ds_load A frags, global B frags, WMMA
// All O used here are multiples of 128, so no tail handling (EXEC all-1s).
// ---------------------------------------------------------------------------
template <int ACT, int K>
__global__ __launch_bounds__(256) void gemm_wmma(
    const float* __restrict__ in, const float* __restrict__ W,
    const float* __restrict__ bias, float* __restrict__ out,
    int O, int tilesN, int colGroups, long inPlane, long outPlane)
{
  constexpr int STRIDE = K + LDS_PAD;            // 16B-aligned, bank-spread rows
  __shared__ __align__(16) float As[16 * STRIDE];

  int blk = blockIdx.x;
  int tn = blk % tilesN;
  int rest = blk / tilesN;
  int cg = rest % colGroups;
  int plane = rest / colGroups;

  const float* aBase = in + plane * inPlane + (long)(tn * 16) * K;

  // ---- cooperative stage of A strip (16 x K floats) into LDS --------------
  {
    const int chunks = 16 * (K / 4);             // 16-byte chunks
    for (int c = threadIdx.x; c < chunks; c += 256) {
      int row = c / (K / 4);
      int col4 = (c % (K / 4)) * 4;
      const float* gp = aBase + (long)row * K + col4;
      float* lp = &As[row * STRIDE + col4];
#if HAVE_ASYNC_LDS
      __builtin_amdgcn_global_load_async_to_lds_b128(
          (gv4i*)gp, (lv4i*)lp, /*offset=*/0, /*cpol=*/0);
#else
      *(float4*)lp = *(const float4*)gp;
#endif
    }
#if HAVE_ASYNC_LDS
#if __has_builtin(__builtin_amdgcn_s_wait_asynccnt)
    __builtin_amdgcn_s_wait_asynccnt(0);
#else
    asm volatile("s_wait_asynccnt 0" ::: "memory");
#endif
#endif
    __syncthreads();
  }

  int lane = threadIdx.x & 31;
  int w = threadIdx.x >> 5;                      // wave id -> col tile in group
  int m = lane & 15;                             // row (A) / col (B,C,D)
  int hi = lane >> 4;                            // K-half / M-half selector

  int to = cg * 8 + w;
  const float* bRow = W + (long)(to * 16 + m) * K;   // W is (O,K): row = out ch
  const float* aLds = &As[m * STRIDE];
  __builtin_prefetch(bRow, 0, 3);

  v8f acc = {0.f, 0.f, 0.f, 0.f, 0.f, 0.f, 0.f, 0.f};
#pragma unroll
  for (int kk = 0; kk < K; kk += 4) {
    // A frag: lane holds A[m][kk+2*hi+{0,1}]  (ISA 16x4 fp32 layout)
    v2f a = *(const v2f*)(aLds + kk + 2 * hi);
    // B frag: lane holds B[kk+2*hi+{0,1}][m] = W[m-th out ch][k]
    v2f b = *(const v2f*)(bRow + kk + 2 * hi);
    acc = __builtin_amdgcn_wmma_f32_16x16x4_f32(
        /*neg_a=*/false, a, /*neg_b=*/false, b,
        /*c_mod=*/(short)0, acc, /*reuse_a=*/false, /*reuse_b=*/false);
  }

  int col = to * 16 + m;
  float bv = bias[col];
  float* oP = out + plane * outPlane;
#pragma unroll
  for (int r = 0; r < 8; ++r) {
    int row = tn * 16 + hi * 8 + r;              // C/D layout: VGPR r -> M=r(+8)
    float x = acc[r] + bv;
    if (ACT) x = x / (1.f + __expf(-x));         // SiLU
    oP[(long)row * O + col] = x;
  }
}

// ---------------------------------------------------------------------------
// Per-edge message: RBF -> K=20 linear -> cosine cutoff envelope,
// pre = phi[dst]*W, scatter-add ds into s (in place) and dv into vout
// (double buffer: vin is pre-message v, vout starts as its copy).
// One 128-thread block per edge; thread t owns channels t, t+128, t+256.
// ---------------------------------------------------------------------------
__global__ __launch_bounds__(128) void edge_message(
    const float* __restrict__ pos, const int* __restrict__ esrc,
    const int* __restrict__ edst, const float* __restrict__ phi,
    const float* __restrict__ vin, float* __restrict__ vout,
    float* __restrict__ s, const float* __restrict__ rw,
    const float* __restrict__ rb, int E, long NF)
{
  int e = blockIdx.x;
  if (e >= E) return;
  int t = threadIdx.x;

  __shared__ float sh_rbf[NRBF];
  __shared__ float sh_r[3];
  __shared__ float sh_norm;

  if (t == 0) {
    int i = esrc[e], j = edst[e];
    float rx = pos[3 * j + 0] - pos[3 * i + 0];
    float ry = pos[3 * j + 1] - pos[3 * i + 1];
    float rz = pos[3 * j + 2] - pos[3 * i + 2];
    sh_r[0] = rx; sh_r[1] = ry; sh_r[2] = rz;
    sh_norm = sqrtf(rx * rx + ry * ry + rz * rz);
  }
  __syncthreads();
  float nrm = sh_norm;
  if (t < NRBF) {
    float nr = fmaxf(nrm, EPS);
    float val = sinf((float)(t + 1) * (PI_F / CUTOFF) * nr) / (nr + EPS);
    sh_rbf[t] = fminf(fmaxf(val, -1.f), 1.f);
  }
  __syncthreads();

  int j = edst[e];
  float pre[3];
#pragma unroll
  for (int p = 0; p < 3; ++p) {
    int c = t + p * F;
    float x = rb[c];
    const float* wr = rw + (long)c * NRBF;
#pragma unroll
    for (int q = 0; q < NRBF; ++q) x += wr[q] * sh_rbf[q];
    float Wv = (x < CUTOFF) ? 0.5f * (__cosf(x * (PI_F / CUTOFF)) + 1.f) : 0.f;
    pre[p] = phi[(long)j * (3 * F) + c] * Wv;
  }

  atomicAdd(&s[(long)j * F + t], pre[1]);                 // ds scatter
  float inv2 = 1.f / ((nrm + EPS) * (nrm + EPS));         // r_hat/(norm+eps)
#pragma unroll
  for (int k = 0; k < 3; ++k) {
    long idx = k * NF + (long)j * F + t;
    float left = vin[idx] * pre[0] + pre[2] * sh_r[k] * inv2;
    atomicAdd(&vout[idx], left);                          // dv scatter
  }
}

// h = concat(||Vv||_k, s)
__global__ void make_h(const float* __restrict__ Vv, const float* __restrict__ s,
                       float* __restrict__ h, int N, long NF)
{
  long i = (long)blockIdx.x * blockDim.x + threadIdx.x;
  if (i >= (long)N * F) return;
  long n = i / F; int g = (int)(i % F);
  float a = Vv[i], b = Vv[NF + i], c = Vv[2 * NF + i];
  h[n * (2 * F) + g] = sqrtf(a * a + b * b + c * c);
  h[n * (2 * F) + F + g] = s[i];
}

// v += a_vv*Uv ; s += sum_k(Uv*Vv)*a_sv + a_ss
__global__ void apply_update(float* __restrict__ v, float* __restrict__ s,
                             const float* __restrict__ Uv,
                             const float* __restrict__ Vv,
                             const float* __restrict__ m, int N, long NF)
{
  long i = (long)blockIdx.x * blockDim.x + threadIdx.x;
  if (i >= (long)N * F) return;
  long n = i / F; int g = (int)(i % F);
  const float* mr = m + n * (3 * F);
  float avv = mr[g], asv = mr[F + g], ass = mr[2 * F + g];
  float dot = 0.f;
#pragma unroll
  for (int k = 0; k < 3; ++k) {
    float u = Uv[k * NF + i], w = Vv[k * NF + i];
    dot += u * w;
    v[k * NF + i] += avv * u;
  }
  s[i] += dot * asv + ass;
}

// s = emb[atoms], v0 = 0
__global__ void init_sv(const int* __restrict__ atoms,
                        const float* __restrict__ emb,
                        float* __restrict__ s, float* __restrict__ v0, int N)
{
  long i = (long)blockIdx.x * blockDim.x + threadIdx.x;
  long nf = (long)N * F;
  if (i >= 3 * nf) return;
  v0[i] = 0.f;
  if (i < nf) {
    long n = i / F; int g = (int)(i % F);
    s[i] = emb[(long)atoms[n] * F + g];
  }
}

// final head O=1: wave-per-node dot product + shuffle reduction
__global__ __launch_bounds__(32) void final_out(
    const float* __restrict__ hid, const float* __restrict__ w2,
    const float* __restrict__ b2, float* __restrict__ out, int N)
{
  int n = blockIdx.x;
  if (n >= N) return;
  int l = threadIdx.x;
  float acc = 0.f;
#pragma unroll
  for (int q = 0; q < 4; ++q)
    acc += hid[(long)n * F + l + q * 32] * w2[l + q * 32];
#pragma unroll
  for (int off = 16; off > 0; off >>= 1) acc += __shfl_xor(acc, off, 32);
  if (l == 0) out[n] = acc + b2[0];
}

extern "C" void kernel_launch(void* const* d_in, const int* in_sizes, int n_in,
                              void* d_out, int out_size, void* d_ws, size_t ws_size,
                              hipStream_t stream)
{
  const int*   atoms   = (const int*)d_in[0];
  const float* pos     = (const float*)d_in[1];
  const int*   esrc    = (const int*)d_in[2];
  const int*   edst    = (const int*)d_in[3];
  const float* emb     = (const float*)d_in[4];
  const float* msg_w1  = (const float*)d_in[5];
  const float* msg_b1  = (const float*)d_in[6];
  const float* msg_w2  = (const float*)d_in[7];
  const float* msg_b2  = (const float*)d_in[8];
  const float* msg_rw  = (const float*)d_in[9];
  const float* msg_rb  = (const float*)d_in[10];
  const float* upd_Uw  = (const float*)d_in[11];
  const float* upd_Ub  = (const float*)d_in[12];
  const float* upd_Vw  = (const float*)d_in[13];
  const float* upd_Vb  = (const float*)d_in[14];
  const float* upd_w1  = (const float*)d_in[15];
  const float* upd_b1  = (const float*)d_in[16];
  const float* upd_w2  = (const float*)d_in[17];
  const float* upd_b2  = (const float*)d_in[18];
  const float* last_w1 = (const float*)d_in[19];
  const float* last_b1 = (const float*)d_in[20];
  const float* last_w2 = (const float*)d_in[21];
  const float* last_b2 = (const float*)d_in[22];

  const int  N  = in_sizes[0];
  const int  E  = in_sizes[2];
  const long NF = (long)N * F;

  float* ws    = (float*)d_ws;
  float* s     = ws;             // (N,F)
  float* vbuf0 = ws + 1 * NF;    // (3,N,F)
  float* vbuf1 = ws + 4 * NF;    // (3,N,F)
  float* phiUv = ws + 7 * NF;    // phi (N,3F) then Uv (3,N,F)
  float* Vv    = ws + 10 * NF;   // (3,N,F)
  float* mbuf  = ws + 13 * NF;   // (N,3F)
  float* hid   = ws + 16 * NF;   // (N,2F): msg hidden / concat h

  float* vb[2] = {vbuf0, vbuf1};
  int cur = 0;

  const int tilesN = N / 16;
  auto launch_gemm = [&](int act, const float* in, const float* W, const float* b,
                         float* out, int K, int O, int planes,
                         long inPlane, long outPlane) {
    int colGroups = O / 128;
    int grid = tilesN * colGroups * planes;
    if (K == 128) {
      if (act) gemm_wmma<1, 128><<<grid, 256, 0, stream>>>(in, W, b, out, O, tilesN, colGroups, inPlane, outPlane);
      else     gemm_wmma<0, 128><<<grid, 256, 0, stream>>>(in, W, b, out, O, tilesN, colGroups, inPlane, outPlane);
    } else {
      if (act) gemm_wmma<1, 256><<<grid, 256, 0, stream>>>(in, W, b, out, O, tilesN, colGroups, inPlane, outPlane);
      else     gemm_wmma<0, 256><<<grid, 256, 0, stream>>>(in, W, b, out, O, tilesN, colGroups, inPlane, outPlane);
    }
  };

  {
    long total = 3 * NF;
    init_sv<<<(int)((total + 255) / 256), 256, 0, stream>>>(atoms, emb, s, vb[0], N);
  }

  for (int l = 0; l < 3; ++l) {
    // message: phi = lin2(silu(lin1(s)))
    launch_gemm(1, s, msg_w1 + (long)l * F * F, msg_b1 + l * F, hid, F, F, 1, 0, 0);
    launch_gemm(0, hid, msg_w2 + (long)l * 3 * F * F, msg_b2 + (long)l * 3 * F,
                phiUv, F, 3 * F, 1, 0, 0);
    // double-buffer v for scatter (read old v[js], accumulate into copy)
    int nxt = 1 - cur;
    (void)hipMemcpyAsync(vb[nxt], vb[cur], (size_t)(3 * NF) * sizeof(float),
                         hipMemcpyDeviceToDevice, stream);
    edge_message<<<E, 128, 0, stream>>>(pos, esrc, edst, phiUv, vb[cur], vb[nxt], s,
                                        msg_rw + (long)l * 3 * F * NRBF,
                                        msg_rb + (long)l * 3 * F, E, NF);
    cur = nxt;
    // update: Uv, Vv projections (3 spatial planes batched)
    launch_gemm(0, vb[cur], upd_Uw + (long)l * F * F, upd_Ub + l * F, phiUv,
                F, F, 3, NF, NF);
    launch_gemm(0, vb[cur], upd_Vw + (long)l * F * F, upd_Vb + l * F, Vv,
                F, F, 3, NF, NF);
    {
      long total = NF;
      make_h<<<(int)((total + 255) / 256), 256, 0, stream>>>(Vv, s, hid, N, NF);
    }
    float* hidden2 = vb[1 - cur];  // dead v buffer doubles as scratch
    launch_gemm(1, hid, upd_w1 + (long)l * F * 2 * F, upd_b1 + l * F, hidden2,
                2 * F, F, 1, 0, 0);
    launch_gemm(0, hidden2, upd_w2 + (long)l * 3 * F * F, upd_b2 + (long)l * 3 * F,
                mbuf, F, 3 * F, 1, 0, 0);
    {
      long total = NF;
      apply_update<<<(int)((total + 255) / 256), 256, 0, stream>>>(
          vb[cur], s, phiUv, Vv, mbuf, N, NF);
    }
  }

  launch_gemm(1, s, last_w1, last_b1, hid, F, F, 1, 0, 0);
  final_out<<<N, 32, 0, stream>>>(hid, last_w2, last_b2, (float*)d_out, N);
}